// TorchNeighborList_4312147165489
// MI455X (gfx1250) — compile-verified
//
#include <hip/hip_runtime.h>

// CDNA5 / gfx1250 neighbor list.
// d^2(i,j) for every 16x16 (i,j) tile is computed with one
// V_WMMA_F32_16X16X4_F32:   D = A(16x4) x B(4x16) + C(16x16)
//   A row m  = [ qx  qy  qz  1 ]            (q = pos_i + case_shift)
//   B col n  = [ -2px -2py -2pz |p|^2 ]^T   (p = pos_j)
//   C[m][n]  = |q_m|^2
// => D[m][n] = |q_m - p_n|^2
// Output writes (~380 MB) dominate: fill pass uses b128 stores only.
// Hit scatter is sparse (~11K of 15.9M candidates): waves with no hit in
// their 16x16 tile exit via a wave-uniform ballot right after the WMMA.

typedef __attribute__((ext_vector_type(2))) float v2f;
typedef __attribute__((ext_vector_type(8))) float v8f;

#define TILE 16
#define MAX_N 1024
#define MAX_S 64

struct NlWs {
  int   count[MAX_N];    // per-atom valid-pair histogram
  int   cursor[MAX_N];   // per-atom placement cursor
  int   start[MAX_N + 1];
  int   total;
  float sv[MAX_S * 3];   // shifts @ box
  float sqp[MAX_N];      // |pos|^2
};

// ---------------------------------------------------------------------------
// Pass 0: shift vectors, |pos|^2, zero the counters (ws is not re-poisoned
// between graph replays, so we must re-init every launch).
__global__ void nl_precompute(const float* __restrict__ pos,
                              const float* __restrict__ box,
                              const int* __restrict__ shifts,
                              int n, int S, NlWs* __restrict__ ws) {
  int t = blockIdx.x * blockDim.x + threadIdx.x;
  if (t < n) {
    float x = pos[3 * t + 0], y = pos[3 * t + 1], z = pos[3 * t + 2];
    ws->sqp[t] = x * x + y * y + z * z;
    ws->count[t] = 0;
    ws->cursor[t] = 0;
  }
  if (t < S) {
    float s0 = (float)shifts[3 * t + 0];
    float s1 = (float)shifts[3 * t + 1];
    float s2 = (float)shifts[3 * t + 2];
#pragma unroll
    for (int d = 0; d < 3; ++d)
      ws->sv[3 * t + d] = s0 * box[0 * 3 + d] + s1 * box[1 * 3 + d] + s2 * box[2 * 3 + d];
  }
}

// ---------------------------------------------------------------------------
// Directed shift case -> effective shift vector v (valid iff |pos_a - pos_b + v| < cutoff)
// Reference emits offset = -v for every directed entry.
__device__ __forceinline__ void nl_case_vec(const NlWs* __restrict__ ws, int cse, int S,
                                            float& vx, float& vy, float& vz) {
  vx = 0.0f; vy = 0.0f; vz = 0.0f;
  if (cse > 0) {
    int   s   = (cse <= S) ? (cse - 1) : (cse - 1 - S);
    float sgn = (cse <= S) ? 1.0f : -1.0f;
    vx = sgn * ws->sv[3 * s + 0];
    vy = sgn * ws->sv[3 * s + 1];
    vz = sgn * ws->sv[3 * s + 2];
  }
}

// ---------------------------------------------------------------------------
// One wave32 computes one 16x16 tile of d^2 with a single f32 WMMA.
// Register layouts per CDNA5 ISA 7.12.2:
//   A 16x4 f32 : lanes 0-15 hold (K0,K1) for M=lane; lanes 16-31 hold (K2,K3) for M=lane-16
//   B 4x16 f32 : mirrored (lanes 0-15 -> rows K0,K1 at N=lane; lanes 16-31 -> rows K2,K3)
//   C/D 16x16  : VGPR v holds (M = v + 8*(lane>=16), N = lane&15)
// C operand (|q_m|^2 per row) is built with wave shuffles (ds_bpermute_b32,
// LDS lane-permute hardware; no extra global loads).
__device__ __forceinline__ v8f nl_tile_d2(const float* __restrict__ pos,
                                          const NlWs* __restrict__ ws,
                                          int i0, int j0,
                                          float vx, float vy, float vz,
                                          int lane) {
  const int half = lane >> 4;   // 0: K0/K1 rows, 1: K2/K3 rows
  const int l    = lane & 15;

  const float* pa = pos + 3 * (i0 + l);
  const float* pb = pos + 3 * (j0 + l);

  // Each lane owns row l of the A tile: q = pos_i + v, and its square.
  float qx = pa[0] + vx, qy = pa[1] + vy, qz = pa[2] + vz;
  float sqq = qx * qx + qy * qy + qz * qz;

  v2f a, b;
  a.x = half ? qz                : qx;
  a.y = half ? 1.0f              : qy;
  b.x = half ? (-2.0f * pb[2])   : (-2.0f * pb[0]);
  b.y = half ? ws->sqp[j0 + l]   : (-2.0f * pb[1]);

  // Broadcast the 8 row-squares this lane's accumulators need.
  v8f c;
#pragma unroll
  for (int v = 0; v < 8; ++v)
    c[v] = __shfl(sqq, half * 8 + v);   // rows 0..7 or 8..15 (lanes 0..15 hold them)

  // D = A*B + C  (8 args: neg_a, A, neg_b, B, c_mod, C, reuse_a, reuse_b)
  return __builtin_amdgcn_wmma_f32_16x16x4_f32(
      /*neg_a=*/false, a, /*neg_b=*/false, b,
      /*c_mod=*/(short)0, c, /*reuse_a=*/false, /*reuse_b=*/false);
}

// Wave-uniform "does this tile contain any candidate hit?" (superset test:
// ignores the diagonal exclusion, which only removes hits).
__device__ __forceinline__ bool nl_tile_any_hit(const v8f& d2, float c2) {
  float m01 = fminf(d2[0], d2[1]), m23 = fminf(d2[2], d2[3]);
  float m45 = fminf(d2[4], d2[5]), m67 = fminf(d2[6], d2[7]);
  float dmin = fminf(fminf(m01, m23), fminf(m45, m67));
  return __any(dmin < c2);
}

// ---------------------------------------------------------------------------
// Pass 1: histogram of valid pairs per idx_i (counting-sort key).
__global__ void nl_count(const float* __restrict__ pos,
                         const float* __restrict__ cut,
                         int n, int S, NlWs* __restrict__ ws) {
  const int ntiles = n / TILE;
  const int tile_i = blockIdx.x * blockDim.y + threadIdx.y;  // uniform per wave
  if (tile_i >= ntiles) return;
  const int tile_j = blockIdx.y;
  const int cse    = blockIdx.z;

  const int i0 = tile_i * TILE, j0 = tile_j * TILE;
  __builtin_prefetch(pos + 3 * i0, 0, 0);   // global_prefetch_b8
  __builtin_prefetch(pos + 3 * j0, 0, 0);

  float vx, vy, vz;
  nl_case_vec(ws, cse, S, vx, vy, vz);
  const float c2 = cut[0] * cut[0];

  const int lane = threadIdx.x;
  v8f d2 = nl_tile_d2(pos, ws, i0, j0, vx, vy, vz, lane);

  if (!nl_tile_any_hit(d2, c2)) return;     // ~99.9% of waves exit here

  const int half = lane >> 4, l = lane & 15;
#pragma unroll
  for (int v = 0; v < 8; ++v) {
    int gi = i0 + half * 8 + v;
    int gj = j0 + l;
    bool ok = (d2[v] < c2) && !(cse == 0 && gi == gj);
    if (ok) atomicAdd(&ws->count[gi], 1);
  }
}

// ---------------------------------------------------------------------------
// Pass 2: exclusive prefix scan over the n buckets (trivial; single lane).
__global__ void nl_scan(NlWs* __restrict__ ws, int n) {
  if (threadIdx.x == 0 && blockIdx.x == 0) {
    int acc = 0;
    for (int i = 0; i < n; ++i) { ws->start[i] = acc; acc += ws->count[i]; }
    ws->start[n] = acc;
    ws->total    = acc;
  }
}

// ---------------------------------------------------------------------------
// Pass 3: stream the invalid-row defaults over the whole output with b128
// stores (store-BW bound: ~380 MB -> ~16 us at 23.3 TB/s; roofline floor).
// Layout [idxi M][idxj M][off 3M][valid M]: first 2M words are -1 (int idx
// default), remaining 4M words are 0 (offset/valid default).
__global__ void nl_fill(uint4* __restrict__ out, long long nvec4, long long split4) {
  long long t = (long long)blockIdx.x * blockDim.x + threadIdx.x;
  if (t >= nvec4) return;
  unsigned w = (t < split4) ? 0xFFFFFFFFu : 0u;
  uint4 q; q.x = w; q.y = w; q.z = w; q.w = w;
  out[t] = q;   // global_store_b128
}

// ---------------------------------------------------------------------------
// Pass 4: recompute tiles and scatter valid pairs into their idx_i bucket
// (valid rows occupy [start[i], start[i]+count[i]) -> globally sorted by idx_i).
__global__ void nl_place(const float* __restrict__ pos,
                         const float* __restrict__ cut,
                         int n, int S, NlWs* __restrict__ ws,
                         int* __restrict__ idxi, int* __restrict__ idxj,
                         float* __restrict__ off, float* __restrict__ valid) {
  const int ntiles = n / TILE;
  const int tile_i = blockIdx.x * blockDim.y + threadIdx.y;
  if (tile_i >= ntiles) return;
  const int tile_j = blockIdx.y;
  const int cse    = blockIdx.z;

  const int i0 = tile_i * TILE, j0 = tile_j * TILE;
  __builtin_prefetch(pos + 3 * i0, 0, 0);
  __builtin_prefetch(pos + 3 * j0, 0, 0);

  float vx, vy, vz;
  nl_case_vec(ws, cse, S, vx, vy, vz);
  const float c2 = cut[0] * cut[0];
  const float ox = -vx, oy = -vy, oz = -vz;  // reference offset = -effective shift

  const int lane = threadIdx.x;
  v8f d2 = nl_tile_d2(pos, ws, i0, j0, vx, vy, vz, lane);

  if (!nl_tile_any_hit(d2, c2)) return;      // ~99.9% of waves exit here

  const int half = lane >> 4, l = lane & 15;
#pragma unroll
  for (int v = 0; v < 8; ++v) {
    int gi = i0 + half * 8 + v;
    int gj = j0 + l;
    bool ok = (d2[v] < c2) && !(cse == 0 && gi == gj);
    if (ok) {
      int slot = ws->start[gi] + atomicAdd(&ws->cursor[gi], 1);
      idxi[slot] = gi;
      idxj[slot] = gj;
      off[3 * slot + 0] = ox;
      off[3 * slot + 1] = oy;
      off[3 * slot + 2] = oz;
      valid[slot] = 1.0f;
    }
  }
}

// ---------------------------------------------------------------------------
extern "C" void kernel_launch(void* const* d_in, const int* in_sizes, int n_in,
                              void* d_out, int out_size, void* d_ws, size_t ws_size,
                              hipStream_t stream) {
  const float* pos    = (const float*)d_in[0];  // (N,3) f32
  const float* box    = (const float*)d_in[1];  // (3,3) f32
  const int*   shifts = (const int*)d_in[2];    // (S,3) i32
  const float* cut    = (const float*)d_in[3];  // scalar f32

  const int n = in_sizes[0] / 3;                // 768
  const int S = in_sizes[2] / 3;                // 13
  const long long M = (long long)n * (n - 1) + 2LL * S * n * n;  // 15,924,480 rows

  NlWs* ws = (NlWs*)d_ws;

  // Output layout (flat, return order): idx_i[M], idx_j[M], offset[M][3], valid[M]
  int*   idxi  = (int*)d_out;
  int*   idxj  = idxi + M;
  float* off   = (float*)(idxj + M);
  float* valid = off + 3 * M;

  const int pre_n = (n > S ? n : S);
  nl_precompute<<<(pre_n + 255) / 256, 256, 0, stream>>>(pos, box, shifts, n, S, ws);

  const int ntiles = n / TILE;                  // 48
  dim3 blk(32, 4);                              // 4 wave32 tiles per block
  dim3 grd((unsigned)((ntiles + 3) / 4), (unsigned)ntiles, (unsigned)(1 + 2 * S));

  nl_count<<<grd, blk, 0, stream>>>(pos, cut, n, S, ws);
  nl_scan<<<1, 32, 0, stream>>>(ws, n);

  // 6M output words total; first 2M words (idx arrays) default to -1, rest to 0.
  const long long nwords = 6LL * M;
  const long long nvec4  = nwords / 4;          // 6M % 4 == 0
  const long long split4 = (2LL * M) / 4;       // 2M % 4 == 0
  nl_fill<<<(unsigned)((nvec4 + 255) / 256), 256, 0, stream>>>((uint4*)d_out, nvec4, split4);

  nl_place<<<grd, blk, 0, stream>>>(pos, cut, n, S, ws, idxi, idxj, off, valid);
}